// TripleBatchTransform_53163105190010
// MI455X (gfx1250) — compile-verified
//
#include <hip/hip_runtime.h>

#define STD_F 0.04f

typedef __attribute__((ext_vector_type(2))) float v2f;
typedef __attribute__((ext_vector_type(8))) float v8f;

// Sum a value across all 32 lanes of a wave using V_WMMA_F32_16X16X4_F32.
// A[m,k]: lane l (<16) supplies A[l,{0,1}] via its 2 A-VGPRs; lane l>=16
// supplies A[l-16,{2,3}]. With a = {s,s} and B = ones(4x16):
//   D[m,n] = 2*(s[m] + s[m+16])  for every column n.
// Each lane holds 8 rows of one column in d[0..7]; adding them and folding
// across the lane-16 half gives 2*sum(all 32 lanes) in every lane.
// Requires EXEC all-ones (call from uniform control flow of a full wave).
__device__ __forceinline__ float wave_sum32_wmma(float s) {
  v2f a = {s, s};
  v2f b = {1.0f, 1.0f};
  v8f c = {};
  v8f d = __builtin_amdgcn_wmma_f32_16x16x4_f32(
      /*neg_a=*/false, a, /*neg_b=*/false, b,
      /*c_mod=*/(short)0, c, /*reuse_a=*/false, /*reuse_b=*/false);
  float r = ((d[0] + d[1]) + (d[2] + d[3])) + ((d[4] + d[5]) + (d[6] + d[7]));
  r += __shfl_xor(r, 16, 32);
  return 0.5f * r;
}

// Block-wide (exactly 1024 threads = 32 waves) reduction of S1,S2 -> mu, 1/sd.
__device__ __forceinline__ void block_stats_1024(float s1, float s2, int half,
                                                 float& mu, float& inv) {
  __shared__ float r1[32];
  __shared__ float r2[32];
  __shared__ float bcast[2];
  const int lane = threadIdx.x & 31;
  const int wid  = threadIdx.x >> 5;

  float w1 = wave_sum32_wmma(s1);
  float w2 = wave_sum32_wmma(s2);
  if (lane == 0) { r1[wid] = w1; r2[wid] = w2; }
  __syncthreads();
  if (wid == 0) {
    float t1 = wave_sum32_wmma(r1[lane]);
    float t2 = wave_sum32_wmma(r2[lane]);
    if (lane == 0) {
      float m = t1 / (float)half;
      float v = t2 / (float)half - m * m;
      v = fmaxf(v, 1e-30f);
      bcast[0] = m;
      bcast[1] = 1.0f / sqrtf(v);
    }
  }
  __syncthreads();
  mu  = bcast[0];
  inv = bcast[1];
}

// One block per row. Single pass over x: pooled values live in registers
// (2*NIT per thread), then a block reduction, then one coalesced b128 store
// pass. Layout: iteration k, thread t owns pooled pairs jA=2t+2048k, jA+1.
template <int NIT>
__global__ __launch_bounds__(1024) void fused_row_kernel(
    const float* __restrict__ x, const float* __restrict__ noise,
    const int* __restrict__ pmove, float* __restrict__ out, int L) {
  const int half = L >> 1;
  int mv = pmove[0] % L;
  if (mv < 0) mv += L;
  const float* xb = x + (size_t)blockIdx.x * (size_t)L;
  float*       ob = out + (size_t)blockIdx.x * (size_t)L;
  const int t = threadIdx.x;

  float pv[2 * NIT];
  float s1 = 0.f, s2 = 0.f;

  if ((mv & 3) == 0) {
    // Shift is a multiple of 4 -> rolled source index stays 16B-aligned and
    // cannot wrap mid-float4 (L % 4 == 0 guaranteed by the launcher).
#pragma unroll
    for (int k = 0; k < NIT; ++k) {
      const int jA = (t << 1) + (k << 11);
      float pA = 0.f, pB = 0.f;
      if (jA < half) {
        const int i0 = jA << 1;              // multiple of 4
        int a = i0 - mv;
        if (a < 0) a += L;                   // multiple of 4, <= L-4
        const float4 xv = *(const float4*)(xb + a);
        const float4 nv = *(const float4*)(noise + i0);
        float z0 = fmaf(nv.x, STD_F, xv.x);
        float z1 = fmaf(nv.y, STD_F, xv.y);
        float z2 = fmaf(nv.z, STD_F, xv.z);
        float z3 = fmaf(nv.w, STD_F, xv.w);
        pA = 0.5f * (z0 + z1);
        pB = 0.5f * (z2 + z3);
        s1 += pA + pB;
        s2 += pA * pA + pB * pB;
      }
      pv[2 * k]     = pA;
      pv[2 * k + 1] = pB;
    }
  } else {
    // Generic shift: scalar loads (still coalesced across the wave).
#pragma unroll
    for (int k = 0; k < NIT; ++k) {
      const int jA = (t << 1) + (k << 11);
      float pA = 0.f, pB = 0.f;
      if (jA < half) {
        const int i0 = jA << 1;
        int a0 = i0     - mv; if (a0 < 0) a0 += L;
        int a1 = i0 + 1 - mv; if (a1 < 0) a1 += L;
        int a2 = i0 + 2 - mv; if (a2 < 0) a2 += L;
        int a3 = i0 + 3 - mv; if (a3 < 0) a3 += L;
        float z0 = fmaf(noise[i0    ], STD_F, xb[a0]);
        float z1 = fmaf(noise[i0 + 1], STD_F, xb[a1]);
        float z2 = fmaf(noise[i0 + 2], STD_F, xb[a2]);
        float z3 = fmaf(noise[i0 + 3], STD_F, xb[a3]);
        pA = 0.5f * (z0 + z1);
        pB = 0.5f * (z2 + z3);
        s1 += pA + pB;
        s2 += pA * pA + pB * pB;
      }
      pv[2 * k]     = pA;
      pv[2 * k + 1] = pB;
    }
  }

  float mu, inv;
  block_stats_1024(s1, s2, half, mu, inv);

#pragma unroll
  for (int k = 0; k < NIT; ++k) {
    const int jA = (t << 1) + (k << 11);
    if (jA < half) {
      float yA = (pv[2 * k]     - mu) * inv;
      float yB = (pv[2 * k + 1] - mu) * inv;
      float4 w;
      w.x = yA; w.y = yA; w.z = yB; w.w = yB;
      *(float4*)(ob + (jA << 1)) = w;  // 16B-aligned, coalesced
    }
  }
}

// Generic fallback (any even L): two streaming passes, same math.
__global__ __launch_bounds__(1024) void generic_row_kernel(
    const float* __restrict__ x, const float* __restrict__ noise,
    const int* __restrict__ pmove, float* __restrict__ out, int L) {
  const int half = L >> 1;
  int mv = pmove[0] % L;
  if (mv < 0) mv += L;
  const float* xb = x + (size_t)blockIdx.x * (size_t)L;
  float*       ob = out + (size_t)blockIdx.x * (size_t)L;

  float s1 = 0.f, s2 = 0.f;
  for (int j = threadIdx.x; j < half; j += 1024) {
    int i0 = j << 1;
    int a0 = i0     - mv; if (a0 < 0) a0 += L;
    int a1 = i0 + 1 - mv; if (a1 < 0) a1 += L;
    float p = 0.5f * (fmaf(noise[i0], STD_F, xb[a0]) +
                      fmaf(noise[i0 + 1], STD_F, xb[a1]));
    s1 += p;
    s2 += p * p;
  }
  float mu, inv;
  block_stats_1024(s1, s2, half, mu, inv);
  for (int j = threadIdx.x; j < half; j += 1024) {
    int i0 = j << 1;
    int a0 = i0     - mv; if (a0 < 0) a0 += L;
    int a1 = i0 + 1 - mv; if (a1 < 0) a1 += L;
    float p = 0.5f * (fmaf(noise[i0], STD_F, xb[a0]) +
                      fmaf(noise[i0 + 1], STD_F, xb[a1]));
    float y = (p - mu) * inv;
    ob[i0]     = y;
    ob[i0 + 1] = y;
  }
}

extern "C" void kernel_launch(void* const* d_in, const int* in_sizes, int n_in,
                              void* d_out, int out_size, void* d_ws,
                              size_t ws_size, hipStream_t stream) {
  (void)n_in; (void)out_size; (void)d_ws; (void)ws_size;
  const float* x     = (const float*)d_in[0];
  const float* noise = (const float*)d_in[1];
  const int*   pmove = (const int*)d_in[2];
  float*       out   = (float*)d_out;

  const int L = in_sizes[1];          // noise length == trace length
  const int B = in_sizes[0] / L;      // rows

  constexpr int NIT = 25;             // covers L/2 <= 25*2048 (L <= 102400)
  if ((L % 4) == 0 && (L >> 1) <= NIT * 2048) {
    hipLaunchKernelGGL((fused_row_kernel<NIT>), dim3(B), dim3(1024), 0, stream,
                       x, noise, pmove, out, L);
  } else {
    hipLaunchKernelGGL(generic_row_kernel, dim3(B), dim3(1024), 0, stream,
                       x, noise, pmove, out, L);
  }
}